// ADMM_24197845745787
// MI455X (gfx1250) — compile-verified
//
#include <hip/hip_runtime.h>

// ADMM LASSO: N=2048 measurements, M=4096 unknowns, 500 iterations.
// ws layout (float indices):
//   [0,4096)      Hty          (constant H^T x)
//   [4096,8192)   s
//   [8192,12288)  v
//   [12288,16384) u
//   [16384,20480) s_new
//   [20480,20608) hpart[128]   (per-block partial squared residuals)
//   [20608]       done flag    (int)

#define M_DIM  4096
#define N_MEAS 2048
#define RHO_C  0.01f
#define BETA_C (0.01f / (2.0f * 12.5f))   // rho/(2*lambda) = 4e-4
#define EPS_C  1e-6f
#define MAX_IT 500
#define WS_HTY   0
#define WS_S     4096
#define WS_V     8192
#define WS_U     12288
#define WS_SNEW  16384
#define WS_HPART 20480
#define WS_FLAG  20608

typedef __attribute__((ext_vector_type(2))) float v2f;
typedef __attribute__((ext_vector_type(8))) float v8f;

// ---- init: Hty = H^T x, zero s/v/u, clear done flag -----------------------
__global__ __launch_bounds__(256)
void admm_init(const float* __restrict__ x, const float* __restrict__ H,
               float* __restrict__ ws) {
    int m = blockIdx.x * 256 + threadIdx.x;          // 0..4095
    float acc = 0.f;
    for (int n = 0; n < N_MEAS; ++n)
        acc += H[(size_t)n * M_DIM + m] * x[n];
    ws[WS_HTY + m]  = acc;
    ws[WS_S + m]    = 0.f;
    ws[WS_V + m]    = 0.f;
    ws[WS_U + m]    = 0.f;
    if (m == 0) ((int*)ws)[WS_FLAG] = 0;
}

// ---- s_new = left_term @ (Hty + rho*(v-u)) via V_WMMA_F32_16X16X4_F32 -----
// Grid: M/16 = 256 blocks, 128 threads (4 waves). Each block owns 16 rows;
// each wave covers a 1024-wide K slice; partials combined through LDS.
__global__ __launch_bounds__(128)
void admm_gemv_s(const float* __restrict__ L, float* __restrict__ ws) {
    if (((const int*)ws)[WS_FLAG]) return;           // converged: frozen
    __shared__ float r[M_DIM];
    __shared__ float part[64];
    const int tid = threadIdx.x;
    // stage right-hand vector in LDS
    for (int i = tid; i < M_DIM; i += 128)
        r[i] = ws[WS_HTY + i] + RHO_C * (ws[WS_V + i] - ws[WS_U + i]);
    __syncthreads();

    const int wave = tid >> 5, lane = tid & 31;
    const int hi   = (lane & 16) ? 1 : 0;            // lanes 16-31 hold K=2,3
    const int row0 = blockIdx.x * 16;
    const int row  = row0 + (lane & 15);
    const float* arow = L + (size_t)row * M_DIM + 2 * hi;

    v8f acc = {};
    const int kbeg = wave * 1024, kend = kbeg + 1024;
    for (int k0 = kbeg; k0 < kend; k0 += 4) {
        v2f a = *(const v2f*)(arow + k0);            // A 16x4: row-streamed
        v2f b;                                       // B 4x16: vector replicated
        b.x = r[k0 + 2 * hi];
        b.y = r[k0 + 2 * hi + 1];
        acc = __builtin_amdgcn_wmma_f32_16x16x4_f32(
            false, a, false, b, (short)0, acc, false, false);
    }
    // D column 0 lives in lane 0 (M=0..7) and lane 16 (M=8..15)
    if (lane == 0) {
#pragma unroll
        for (int g = 0; g < 8; ++g) part[wave * 16 + g] = acc[g];
    } else if (lane == 16) {
#pragma unroll
        for (int g = 0; g < 8; ++g) part[wave * 16 + 8 + g] = acc[g];
    }
    __syncthreads();
    if (tid < 16)
        ws[WS_SNEW + row0 + tid] =
            part[tid] + part[16 + tid] + part[32 + tid] + part[48 + tid];
}

// ---- shrinkage + dual update + deterministic delta reduce + flag ----------
__global__ __launch_bounds__(1024)
void admm_update(float* __restrict__ ws) {
    __shared__ float red[1024];
    int* flag = (int*)ws + WS_FLAG;
    const int tid = threadIdx.x;
    const int done = *flag;
    float local = 0.f;
    if (!done) {
        for (int m = tid; m < M_DIM; m += 1024) {
            float sn = ws[WS_SNEW + m];
            float sp = ws[WS_S + m];
            local += fabsf(sn - sp);
            float z  = sn + ws[WS_U + m];
            float az = fabsf(z) - BETA_C;
            float vn = (az > 0.f) ? copysignf(az, z) : 0.f;
            float un = ws[WS_U + m] + (sn - vn);     // MU = 1.0
            ws[WS_S + m] = sn;
            ws[WS_V + m] = vn;
            ws[WS_U + m] = un;
        }
    }
    red[tid] = local;
    __syncthreads();
    for (int off = 512; off > 0; off >>= 1) {
        if (tid < off) red[tid] += red[tid + off];
        __syncthreads();
    }
    if (tid == 0 && !done && red[0] <= EPS_C) *flag = 1;
}

// ---- per-block partial of sum((H@s - x)^2) via WMMA -----------------------
__global__ __launch_bounds__(128)
void admm_gemv_h(const float* __restrict__ H, const float* __restrict__ x,
                 float* __restrict__ ws) {
    if (((const int*)ws)[WS_FLAG]) return;           // err forced to 0 anyway
    __shared__ float sv[M_DIM];
    __shared__ float part[64];
    __shared__ float red[16];
    const int tid = threadIdx.x;
    for (int i = tid; i < M_DIM; i += 128) sv[i] = ws[WS_S + i];
    __syncthreads();

    const int wave = tid >> 5, lane = tid & 31;
    const int hi   = (lane & 16) ? 1 : 0;
    const int row0 = blockIdx.x * 16;
    const int row  = row0 + (lane & 15);
    const float* arow = H + (size_t)row * M_DIM + 2 * hi;

    v8f acc = {};
    const int kbeg = wave * 1024, kend = kbeg + 1024;
    for (int k0 = kbeg; k0 < kend; k0 += 4) {
        v2f a = *(const v2f*)(arow + k0);
        v2f b;
        b.x = sv[k0 + 2 * hi];
        b.y = sv[k0 + 2 * hi + 1];
        acc = __builtin_amdgcn_wmma_f32_16x16x4_f32(
            false, a, false, b, (short)0, acc, false, false);
    }
    if (lane == 0) {
#pragma unroll
        for (int g = 0; g < 8; ++g) part[wave * 16 + g] = acc[g];
    } else if (lane == 16) {
#pragma unroll
        for (int g = 0; g < 8; ++g) part[wave * 16 + 8 + g] = acc[g];
    }
    __syncthreads();
    if (tid < 16) {
        float y = part[tid] + part[16 + tid] + part[32 + tid] + part[48 + tid];
        float t = y - x[row0 + tid];
        red[tid] = t * t;
    }
    __syncthreads();
    if (tid == 0) {
        float s2 = 0.f;
        for (int i = 0; i < 16; ++i) s2 += red[i];   // fixed order
        ws[WS_HPART + blockIdx.x] = s2;
    }
}

// ---- errors[it] = done ? 0 : sum of 128 partials --------------------------
__global__ __launch_bounds__(128)
void admm_err(const float* __restrict__ ws, float* __restrict__ out, int it) {
    __shared__ float red[128];
    const int tid = threadIdx.x;
    red[tid] = ws[WS_HPART + tid];
    __syncthreads();
    for (int off = 64; off > 0; off >>= 1) {
        if (tid < off) red[tid] += red[tid + off];
        __syncthreads();
    }
    if (tid == 0)
        out[M_DIM + it] = ((const int*)ws)[WS_FLAG] ? 0.f : red[0];
}

// ---- final: copy s into d_out[0..4095] ------------------------------------
__global__ __launch_bounds__(256)
void admm_copy(const float* __restrict__ ws, float* __restrict__ out) {
    int m = blockIdx.x * 256 + threadIdx.x;
    out[m] = ws[WS_S + m];
}

extern "C" void kernel_launch(void* const* d_in, const int* in_sizes, int n_in,
                              void* d_out, int out_size, void* d_ws, size_t ws_size,
                              hipStream_t stream) {
    const float* x = (const float*)d_in[0];          // [2048]
    const float* H = (const float*)d_in[1];          // [2048*4096]
    const float* L = (const float*)d_in[2];          // [4096*4096] left_term
    float* out = (float*)d_out;                      // [4096 s | 500 errors]
    float* ws  = (float*)d_ws;

    admm_init<<<16, 256, 0, stream>>>(x, H, ws);
    for (int it = 0; it < MAX_IT; ++it) {
        admm_gemv_s<<<M_DIM / 16, 128, 0, stream>>>(L, ws);
        admm_update<<<1, 1024, 0, stream>>>(ws);
        admm_gemv_h<<<N_MEAS / 16, 128, 0, stream>>>(H, x, ws);
        admm_err<<<1, 128, 0, stream>>>(ws, out, it);
    }
    admm_copy<<<16, 256, 0, stream>>>(ws, out);
}